// NNConvNet_81939386073494
// MI455X (gfx1250) — compile-verified
//
#include <hip/hip_runtime.h>

// NNConv GNN forward for MI455X (gfx1250, wave32, WMMA).
//
// Roofline: dominant op is the layer-2 bilinear GEMM (34.4 GFLOPs fp32),
// executed with V_WMMA_F32_16X16X4_F32 on A-fragments generated in-register
// (rank-1 products), so the 1 GiB per-edge-weight intermediate never exists.
// HBM traffic ~50MB (~2us @ 23.3 TB/s); B matrix (1MB) is L2-resident and
// staged through LDS in 64KB slices shared by 8 waves/WG.

#define N_NODES 32768
#define N_EDGES 65536

typedef __attribute__((ext_vector_type(2))) float v2f;
typedef __attribute__((ext_vector_type(8))) float v8f;

__device__ __forceinline__ float elu1(float v) { return v > 0.f ? v : expm1f(v); }

__global__ void zero_kernel(float* __restrict__ p, int n) {
  int i = blockIdx.x * 256 + threadIdx.x;
  if (i < n) p[i] = 0.f;
}

__global__ void cnt_kernel(const int* __restrict__ dstp, float* __restrict__ cnt) {
  int e = blockIdx.x * 256 + threadIdx.x;
  if (e < N_EDGES) atomicAdd(&cnt[dstp[e]], 1.0f);
}

// Bq[o][i*64+k] = Wb[k][i*64+o]  (bilinear-permuted, transposed for LDS rows)
// Bb[o][i]      = bvec[i*64+o]   (zero-padded to bias_k columns)
__global__ void permute_kernel(const float* __restrict__ Wb, const float* __restrict__ bvec,
                               float* __restrict__ Bq, float* __restrict__ Bb,
                               int in_c, int bias_k) {
  int idx = blockIdx.x * 256 + threadIdx.x;
  int K = in_c * 64;
  if (idx < 64 * K) {
    int o = idx / K, k = idx - o * K;
    int i = k >> 6, kk = k & 63;
    Bq[idx] = Wb[kk * K + i * 64 + o];
  }
  if (idx < 64 * bias_k) {
    int o = idx / bias_k, i = idx - o * bias_k;
    Bb[idx] = (i < in_c) ? bvec[i * 64 + o] : 0.f;
  }
}

// Fused: h = relu(ea@Wa+ba); msg = (x_src (x) h) @ Bq + x_src @ Bb; atomic scatter.
// 256 thr = 8 waves; 128 edges per WG; 16 edges (M) per wave; 4 N-tiles (64 out).
template <int IN_C>
__launch_bounds__(256, 1)
__global__ void edge_wmma_kernel(const float* __restrict__ ea,
                                 const int* __restrict__ srcp,
                                 const int* __restrict__ dstp,
                                 const float* __restrict__ xsrc,
                                 const float* __restrict__ Wa,
                                 const float* __restrict__ ba,
                                 const float* __restrict__ Bq,
                                 const float* __restrict__ Bb,
                                 float* __restrict__ summed) {
  constexpr int ISLICE = (IN_C == 3) ? 3 : 4;   // i-values per staged K-slice
  constexpr int NSLICE = IN_C / ISLICE;         // 1 or 16 slices
  constexpr int SK     = ISLICE * 64;           // K per slice: 192 / 256
  constexpr int BROW   = SK + 4;                // row stride: %64==4 -> conflict-free b64
  constexpr int BIAS_K = (IN_C == 3) ? 4 : IN_C;
  constexpr int XROW   = BIAS_K + 4;
  constexpr int BBROW  = BIAS_K + 4;

  __shared__ __align__(16) float Bt[64 * BROW];    // staged B slice   (65KB / 49KB)
  __shared__ __align__(16) float Hh[128 * 68];     // edge-MLP hidden  (34KB)
  __shared__ __align__(16) float Xs[128 * XROW];   // gathered x[src]  (34KB / 4KB)
  __shared__ __align__(16) float Bbt[64 * BBROW];  // permuted bias    (17KB / 2KB)

  const int tid = threadIdx.x;
  const int e0  = blockIdx.x * 128;

  // Stage bias matrix
  for (int idx = tid; idx < 64 * BIAS_K; idx += 256) {
    int o = idx / BIAS_K, k = idx - o * BIAS_K;
    Bbt[o * BBROW + k] = Bb[idx];
  }

  // h = relu(ea @ Wa + ba), 128 edges x 64 hidden (each thread: half a row)
  {
    int el = tid >> 1, kb = (tid & 1) * 32, e = e0 + el;
    float a0 = ea[e * 3 + 0], a1 = ea[e * 3 + 1], a2 = ea[e * 3 + 2];
    #pragma unroll
    for (int k = 0; k < 32; ++k) {
      int kk = kb + k;
      float hv = fmaf(a0, Wa[kk], fmaf(a1, Wa[64 + kk], fmaf(a2, Wa[128 + kk], ba[kk])));
      Hh[el * 68 + kk] = fmaxf(hv, 0.f);
    }
  }

  // Gather source-node features
  {
    int el = tid >> 1, part = tid & 1;
    int s = srcp[e0 + el];
    if (IN_C == 64) {
      const float4* srow = (const float4*)&xsrc[(size_t)s * 64 + part * 32];
      float4* drow = (float4*)&Xs[el * XROW + part * 32];
      #pragma unroll
      for (int j = 0; j < 8; ++j) drow[j] = srow[j];
    } else if (part == 0) {
      Xs[el * XROW + 0] = xsrc[(size_t)s * 3 + 0];
      Xs[el * XROW + 1] = xsrc[(size_t)s * 3 + 1];
      Xs[el * XROW + 2] = xsrc[(size_t)s * 3 + 2];
      Xs[el * XROW + 3] = 0.f;  // pad for bias K-loop
    }
  }
  __syncthreads();

  const int lane = tid & 31;
  const int wv   = tid >> 5;
  const int m    = lane & 15;        // A row / B,C,D column within tile
  const int off  = (lane >> 4) * 2;  // K sub-offset per 16x16x4 layout
  const int erow = wv * 16 + m;
  const float* hrow = &Hh[erow * 68];
  const float* xrow = &Xs[erow * XROW];

  v8f acc[4] = {};

  // Bias term: msg += x_src @ Bb (small WMMA K-loop)
  #pragma unroll
  for (int kb = 0; kb < BIAS_K; kb += 4) {
    v2f a = *(const v2f*)&xrow[kb + off];
    #pragma unroll
    for (int nt = 0; nt < 4; ++nt) {
      v2f b = *(const v2f*)&Bbt[(nt * 16 + m) * BBROW + kb + off];
      acc[nt] = __builtin_amdgcn_wmma_f32_16x16x4_f32(false, a, false, b, (short)0,
                                                      acc[nt], false, false);
    }
  }

  // Main bilinear GEMM: A[m, i*64+k] = x_src[m,i] * h[m,k], built in-register
  for (int s = 0; s < NSLICE; ++s) {
    __syncthreads();
    for (int idx = tid; idx < 64 * (SK / 4); idx += 256) {
      int o = idx / (SK / 4), c4 = idx - o * (SK / 4);
      *(float4*)&Bt[o * BROW + c4 * 4] =
          *(const float4*)&Bq[(size_t)o * (IN_C * 64) + s * SK + c4 * 4];
    }
    __syncthreads();
    #pragma unroll
    for (int il = 0; il < ISLICE; ++il) {
      float xsi = xrow[s * ISLICE + il];
      const float* brow = &Bt[il * 64];
      #pragma unroll
      for (int kk = 0; kk < 64; kk += 4) {
        v2f hp = *(const v2f*)&hrow[kk + off];
        v2f a  = hp * xsi;  // rank-1 A fragment
        #pragma unroll
        for (int nt = 0; nt < 4; ++nt) {
          v2f b = *(const v2f*)&brow[(nt * 16 + m) * BROW + kk + off];
          acc[nt] = __builtin_amdgcn_wmma_f32_16x16x4_f32(false, a, false, b, (short)0,
                                                          acc[nt], false, false);
        }
      }
    }
  }

  // Scatter-add into destination nodes. C/D layout: lane holds N=m, M=j+8*(lane/16).
  int dd[8];
  #pragma unroll
  for (int j = 0; j < 8; ++j) dd[j] = dstp[e0 + wv * 16 + (off >> 1) * 8 + j];
  #pragma unroll
  for (int nt = 0; nt < 4; ++nt) {
    #pragma unroll
    for (int j = 0; j < 8; ++j)
      atomicAdd(&summed[(size_t)dd[j] * 64 + nt * 16 + m], acc[nt][j]);
  }
}

// out = elu(summed/max(cnt,1) + xin @ root + bias); 4 nodes per block.
__global__ void node_update_kernel(const float* __restrict__ summed, const float* __restrict__ cnt,
                                   const float* __restrict__ xin, int in_c,
                                   const float* __restrict__ root, const float* __restrict__ bias,
                                   float* __restrict__ outp) {
  __shared__ float rootS[64 * 64];
  __shared__ float xS[4 * 64];
  int tid = threadIdx.x;
  int n0 = blockIdx.x * 4;
  for (int idx = tid; idx < in_c * 64; idx += 256) rootS[idx] = root[idx];
  for (int idx = tid; idx < 4 * in_c; idx += 256) {
    int nl = idx / in_c, i = idx - nl * in_c;
    xS[nl * 64 + i] = xin[(size_t)(n0 + nl) * in_c + i];
  }
  __syncthreads();
  int nl = tid >> 6, o = tid & 63;
  int n = n0 + nl;
  float c = fmaxf(cnt[n], 1.f);
  float v = summed[(size_t)n * 64 + o] / c + bias[o];
  for (int i = 0; i < in_c; ++i) v = fmaf(xS[nl * 64 + i], rootS[i * 64 + o], v);
  outp[(size_t)n * 64 + o] = elu1(v);
}

// out = elu(h @ Wf + bf); 16 nodes per block, Wf staged in LDS.
__global__ void fc_kernel(const float* __restrict__ h, const float* __restrict__ Wf,
                          const float* __restrict__ bf, float* __restrict__ outp) {
  __shared__ float Ws[64 * 128];
  __shared__ float hS[16 * 64];
  int tid = threadIdx.x;
  int n0 = blockIdx.x * 16;
  for (int idx = tid; idx < 64 * 128; idx += 256) Ws[idx] = Wf[idx];
  for (int idx = tid; idx < 16 * 64; idx += 256) hS[idx] = h[(size_t)n0 * 64 + idx];
  __syncthreads();
  int nl = tid & 15, ob = (tid >> 4) * 8;
  float acc[8];
  #pragma unroll
  for (int j = 0; j < 8; ++j) acc[j] = bf[ob + j];
  for (int k = 0; k < 64; ++k) {
    float hv = hS[nl * 64 + k];
    #pragma unroll
    for (int j = 0; j < 8; ++j) acc[j] = fmaf(hv, Ws[k * 128 + ob + j], acc[j]);
  }
  #pragma unroll
  for (int j = 0; j < 8; ++j) outp[(size_t)(n0 + nl) * 128 + ob + j] = elu1(acc[j]);
}

extern "C" void kernel_launch(void* const* d_in, const int* in_sizes, int n_in,
                              void* d_out, int out_size, void* d_ws, size_t ws_size,
                              hipStream_t stream) {
  const float* x     = (const float*)d_in[0];
  const int*   ei    = (const int*)d_in[1];    // [2, E] int32
  const float* ea    = (const float*)d_in[2];
  const float* W1a   = (const float*)d_in[3];
  const float* b1a   = (const float*)d_in[4];
  const float* W1b   = (const float*)d_in[5];
  const float* b1b   = (const float*)d_in[6];
  const float* root1 = (const float*)d_in[7];
  const float* bias1 = (const float*)d_in[8];
  const float* W2a   = (const float*)d_in[9];
  const float* b2a   = (const float*)d_in[10];
  const float* W2b   = (const float*)d_in[11];
  const float* b2b   = (const float*)d_in[12];
  const float* root2 = (const float*)d_in[13];
  const float* bias2 = (const float*)d_in[14];
  const float* Wf    = (const float*)d_in[15];
  const float* bf    = (const float*)d_in[16];
  float* out = (float*)d_out;

  const int* srcp = ei;
  const int* dstp = ei + N_EDGES;

  // Workspace layout (floats), ~26.5 MB total
  float* w      = (float*)d_ws;
  float* summed = w;                                  // N*64 (reused both layers)
  float* cnt    = summed + (size_t)N_NODES * 64;      // N
  float* x1     = cnt + N_NODES;                      // N*64
  float* h2     = x1 + (size_t)N_NODES * 64;          // N*64
  float* Bq2    = h2 + (size_t)N_NODES * 64;          // 64*4096
  float* Bq1    = Bq2 + 64 * 4096;                    // 64*192
  float* Bb2    = Bq1 + 64 * 192;                     // 64*64
  float* Bb1    = Bb2 + 64 * 64;                      // 64*4

  const int nzero1 = N_NODES * 64 + N_NODES;  // summed + cnt (contiguous)
  zero_kernel<<<(nzero1 + 255) / 256, 256, 0, stream>>>(summed, nzero1);
  permute_kernel<<<(64 * 192 + 255) / 256, 256, 0, stream>>>(W1b, b1b, Bq1, Bb1, 3, 4);
  permute_kernel<<<(64 * 4096 + 255) / 256, 256, 0, stream>>>(W2b, b2b, Bq2, Bb2, 64, 64);
  cnt_kernel<<<N_EDGES / 256, 256, 0, stream>>>(dstp, cnt);

  // Layer 1
  edge_wmma_kernel<3><<<N_EDGES / 128, 256, 0, stream>>>(ea, srcp, dstp, x, W1a, b1a,
                                                         Bq1, Bb1, summed);
  node_update_kernel<<<N_NODES / 4, 256, 0, stream>>>(summed, cnt, x, 3, root1, bias1, x1);

  // Layer 2
  zero_kernel<<<(N_NODES * 64 + 255) / 256, 256, 0, stream>>>(summed, N_NODES * 64);
  edge_wmma_kernel<64><<<N_EDGES / 128, 256, 0, stream>>>(ea, srcp, dstp, x1, W2a, b2a,
                                                          Bq2, Bb2, summed);
  node_update_kernel<<<N_NODES / 4, 256, 0, stream>>>(summed, cnt, x1, 64, root2, bias2, h2);

  // FC + ELU
  fc_kernel<<<N_NODES / 16, 256, 0, stream>>>(h2, Wf, bf, out);
}